// Spectral_corr_mining_89756226552173
// MI455X (gfx1250) — compile-verified
//
#include <hip/hip_runtime.h>
#include <hip/hip_bf16.h>
#include <stdint.h>

typedef __attribute__((ext_vector_type(2))) float v2f;
typedef __attribute__((ext_vector_type(8))) float v8f;
typedef __attribute__((ext_vector_type(4))) unsigned int v4u;
typedef __attribute__((ext_vector_type(8))) int v8i;
typedef __attribute__((ext_vector_type(4))) int v4i;

#define C_DIM 200
#define HW 1024
#define CT 13            // 13 tiles of 16 -> 208 rows/cols
#define CPAD 208
#define KC 32            // K-chunk (columns staged in LDS per step)
#define LDS_STRIDE 34    // 32 dwords + 2 pad dwords -> bank-conflict-free
#define NCHUNK (HW / KC) // 32
#define NWAVES 13
#define NTHREADS (NWAVES * 32)

#if __has_builtin(__builtin_amdgcn_tensor_load_to_lds)
#define USE_TDM 1
#else
#define USE_TDM 0
#endif

// Issue one TDM descriptor: load a (KC x C_DIM) fp32 tile (rows 200..207
// zero-filled via tensor_dim1 OOB) into LDS with a 2-dword pad after every
// 32 dwords (=> LDS row stride of 34 dwords).
__device__ __forceinline__ void tdm_load_chunk(uint64_t gaddr, uint32_t lds_addr) {
#if USE_TDM
  v4u g0;
  g0.x = 1u;                                   // count=1, user descriptor
  g0.y = lds_addr;                             // lds_addr (bytes)
  g0.z = (uint32_t)gaddr;                      // global_addr[31:0]
  g0.w = ((uint32_t)(gaddr >> 32) & 0x01FFFFFFu) | 0x80000000u; // addr[56:32] | type=2

  v8i g1;
  // wg_mask=0 | data_size=2 (4B) | pad_enable | pad_interval=4 (32 dw) | pad_amount=1 (2 dw)
  g1[0] = (int)((2u << 16) | (1u << 20) | (4u << 22) | (1u << 25));
  g1[1] = (int)(((uint32_t)HW & 0xFFFFu) << 16);     // tensor_dim0 lo16 @ bits 63:48
  g1[2] = (int)(((uint32_t)C_DIM & 0xFFFFu) << 16);  // tensor_dim0 hi=0 | tensor_dim1 lo16
  g1[3] = (int)(((uint32_t)KC & 0xFFFFu) << 16);     // tensor_dim1 hi=0 | tile_dim0=32
  g1[4] = (int)(CPAD);                               // tile_dim1=208 | tile_dim2=0
  g1[5] = (int)(HW);                                 // tensor_dim0_stride lo32 = 1024
  g1[6] = 0;                                         // stride hi | tensor_dim1_stride lo
  g1[7] = 0;

  v4i g2;                                            // dims 2/3 = 1 (inactive but valid)
  g2[0] = 1; g2[1] = 1; g2[2] = 0; g2[3] = 0;
  v4i g3;
  g3[0] = 0; g3[1] = (1 << 16); g3[2] = 0; g3[3] = 0; // tensor_dim4 = 1

  v8i g4 = {};                                       // extra 256-bit group (clang-23 form)
  __builtin_amdgcn_tensor_load_to_lds(g0, g1, g2, g3, g4, 0);
#else
  (void)gaddr; (void)lds_addr;
#endif
}

__global__ __launch_bounds__(NTHREADS) void
spectral_corr_mining_kernel(const float* __restrict__ x,
                            const float* __restrict__ temp,
                            float* __restrict__ out) {
  __shared__ float sA[2][CPAD * LDS_STRIDE];  // 2 x 28,288 B, double buffered
  __shared__ float s_d[CPAD];
  __shared__ float s_rowsum[CPAD];
  __shared__ float s_colsum[CPAD];
  __shared__ float s_tot;

  const int tid  = threadIdx.x;
  const int lane = tid & 31;
  const int wave = tid >> 5;          // 0..12, owns output tile-row `wave`
  const int half = lane >> 4;         // 0 or 1
  const int nloc = lane & 15;
  const int b    = blockIdx.x;

  if (tid < CPAD) s_colsum[tid] = 0.0f;
  // Zero pad rows 200..207 of both staging buffers (TDM OOB also zero-fills;
  // required for the non-TDM fallback path).
  for (int i = tid; i < 8 * LDS_STRIDE; i += NTHREADS) {
    sA[0][C_DIM * LDS_STRIDE + i] = 0.0f;
    sA[1][C_DIM * LDS_STRIDE + i] = 0.0f;
  }

  const uint64_t gbase = (uint64_t)(uintptr_t)x +
                         (uint64_t)b * ((uint64_t)C_DIM * HW * 4);
  const uint32_t lds0 = (uint32_t)(uintptr_t)&sA[0][0];
  const uint32_t lds1 = (uint32_t)(uintptr_t)&sA[1][0];

  v8f acc[CT] = {};   // 13 tiles x 16x16 f32 accumulators

#if USE_TDM
  if (wave == 0) tdm_load_chunk(gbase, lds0);
#endif

  for (int ch = 0; ch < NCHUNK; ++ch) {
#if USE_TDM
    if (wave == 0) __builtin_amdgcn_s_wait_tensorcnt(0);
    __syncthreads();  // buf[ch&1] ready; everyone done reading buf[(ch+1)&1]
    if (wave == 0 && ch + 1 < NCHUNK)
      tdm_load_chunk(gbase + (uint64_t)(ch + 1) * (KC * 4),
                     ((ch + 1) & 1) ? lds1 : lds0);
#else
    __syncthreads();
    {
      float* dst = sA[ch & 1];
      const float* src = x + (size_t)b * C_DIM * HW + (size_t)ch * KC;
      for (int i = tid; i < C_DIM * (KC / 2); i += NTHREADS) {
        const int row = i / (KC / 2);
        const int kk  = (i % (KC / 2)) * 2;
        v2f v = *(const v2f*)(src + (size_t)row * HW + kk);
        *(v2f*)(dst + row * LDS_STRIDE + kk) = v;
      }
    }
    __syncthreads();
#endif
    const float* sbuf = sA[ch & 1];

#pragma unroll
    for (int ks = 0; ks < KC / 4; ++ks) {
      const int koff = ks * 4 + 2 * half;  // lanes 16..31 take K+2,K+3
      // A fragment (16x4) for tile-row `wave`; identical layout serves as the
      // B fragment (4x16) of X^T for the Gram product.
      v2f af = *(const v2f*)(sbuf + (wave * 16 + nloc) * LDS_STRIDE + koff);
#pragma unroll
      for (int c = 0; c < CT; ++c) {
        v2f bf = *(const v2f*)(sbuf + (c * 16 + nloc) * LDS_STRIDE + koff);
        acc[c] = __builtin_amdgcn_wmma_f32_16x16x4_f32(
            false, af, false, bf, (short)0, acc[c], false, false);
      }
    }
  }

  // ---- diagonal d[i] = G[i][i] (from the (w,w) tile) ----
#pragma unroll
  for (int c = 0; c < CT; ++c) {
    if (c == wave) {
#pragma unroll
      for (int v = 0; v < 8; ++v) {
        const int m = v + 8 * half;     // local M of this vgpr/lane-half
        if (m == nloc) s_d[wave * 16 + m] = acc[c][v];
      }
    }
  }
  __syncthreads();

  // ---- dcov + row/col partial sums ----
  const float eT = expf(temp[0]);
  const bool validM = !(wave == CT - 1 && half == 1);  // rows 200..207
  float dm[8], rl[8];
#pragma unroll
  for (int v = 0; v < 8; ++v) {
    dm[v] = s_d[wave * 16 + v + 8 * half];
    rl[v] = 0.0f;
  }
#pragma unroll
  for (int c = 0; c < CT; ++c) {
    const float dn = s_d[c * 16 + nloc];
    const bool validN = !(c == CT - 1 && nloc >= 8);   // cols 200..207
    const bool valid = validM && validN;
    float cp = 0.0f;
#pragma unroll
    for (int v = 0; v < 8; ++v) {
      const float t  = fmaxf(dm[v] + dn - 2.0f * acc[c][v], 0.0f);
      const float dc = valid ? sqrtf(fmaf(eT, t, 1e-5f)) : 0.0f;
      acc[c][v] = dc;   // overwrite G tile with dcov
      rl[v] += dc;
      cp    += dc;
    }
    cp += __shfl_xor(cp, 16, 32);       // fold upper-half rows into lower
    if (half == 0) atomicAdd(&s_colsum[c * 16 + nloc], cp);
  }
#pragma unroll
  for (int v = 0; v < 8; ++v) {         // reduce over the 16 lanes of a half
    float r = rl[v];
    r += __shfl_xor(r, 1, 32);
    r += __shfl_xor(r, 2, 32);
    r += __shfl_xor(r, 4, 32);
    r += __shfl_xor(r, 8, 32);
    if (nloc == 0) s_rowsum[wave * 16 + v + 8 * half] = r;
  }
  __syncthreads();

  if (wave == 0) {                      // total sum
    float t = 0.0f;
    for (int j = lane; j < CPAD; j += 32) t += s_colsum[j];
    for (int m = 16; m > 0; m >>= 1) t += __shfl_xor(t, m, 32);
    if (lane == 0) s_tot = t;
  }
  __syncthreads();

  // ---- out = rowmean + colmean - totmean - dcov ----
  const float tm = s_tot * (1.0f / (200.0f * 200.0f));
  float rm[8];
#pragma unroll
  for (int v = 0; v < 8; ++v)
    rm[v] = s_rowsum[wave * 16 + v + 8 * half] * (1.0f / 200.0f);

  float* outB = out + (size_t)b * (C_DIM * C_DIM);
#pragma unroll
  for (int c = 0; c < CT; ++c) {
    const int N = c * 16 + nloc;
    const float cm = s_colsum[N] * (1.0f / 200.0f);
#pragma unroll
    for (int v = 0; v < 8; ++v) {
      const int M = wave * 16 + v + 8 * half;
      if (M < C_DIM && N < C_DIM)
        outB[(size_t)M * C_DIM + N] = rm[v] + cm - tm - acc[c][v];
    }
  }
}

extern "C" void kernel_launch(void* const* d_in, const int* in_sizes, int n_in,
                              void* d_out, int out_size, void* d_ws, size_t ws_size,
                              hipStream_t stream) {
  const float* x    = (const float*)d_in[0];   // (256, 200, 32, 32) fp32
  const float* temp = (const float*)d_in[1];   // (1, 1) fp32
  float* out        = (float*)d_out;           // (256, 200, 200) fp32
  (void)in_sizes; (void)n_in; (void)out_size; (void)d_ws; (void)ws_size;

  dim3 grid(256);        // one workgroup per batch element
  dim3 block(NTHREADS);  // 13 waves of 32
  spectral_corr_mining_kernel<<<grid, block, 0, stream>>>(x, temp, out);
}